// MultiHeadAttentionLayerGritSparse_59596966199891
// MI455X (gfx1250) — compile-verified
//
#include <hip/hip_runtime.h>
#include <hip/hip_bf16.h>

typedef __attribute__((ext_vector_type(2))) float v2f;
typedef __attribute__((ext_vector_type(8))) float v8f;

#define HEADS   8
#define OUT_DIM 8
#define HD      64      // HEADS*OUT_DIM
#define IN_DIM  64
#define EHD     128     // 2*HD
#define CLAMP_V 5.0f
#define EPS_V   1e-8f

#define NODE_TILES_PER_BLOCK 5   // 3125 = 5 * 625
#define EDGE_TILES_PER_BLOCK 4   // 50000 = 4 * 12500

// ---------------- helpers ----------------
__device__ __forceinline__ void atomic_max_float(float* addr, float val) {
    // sign-magnitude trick: valid for any IEEE float mix, init must be -inf
    if (val >= 0.0f)
        atomicMax((int*)addr, __float_as_int(val));
    else
        atomicMin((unsigned int*)addr, __float_as_uint(val));
}

// ---------------- kernel 0: init accumulators ----------------
__global__ void k_init(float* __restrict__ wV, float* __restrict__ m_arr,
                       float* __restrict__ denom, float* __restrict__ rowV,
                       int n_nodes) {
    int tid = blockIdx.x * blockDim.x + threadIdx.x;
    if (tid < n_nodes * HD) { wV[tid] = 0.0f; rowV[tid] = 0.0f; }
    if (tid < n_nodes * HEADS) { m_arr[tid] = -__builtin_inff(); denom[tid] = 0.0f; }
}

// ---------------- kernel 1: node projections (fp32 WMMA, B in registers) ----------------
// grid = (n_mtiles/5, 3 [Q,K,V]); block = 128 (4 waves, one 16-col tile each)
__global__ void k_node_gemm(const float* __restrict__ x,
                            const float* __restrict__ Wq, const float* __restrict__ bq,
                            const float* __restrict__ Wk, const float* __restrict__ bk,
                            const float* __restrict__ Wv, const float* __restrict__ bv,
                            float* __restrict__ Q, float* __restrict__ K,
                            float* __restrict__ V) {
    const int lane  = threadIdx.x & 31;
    const int wave  = threadIdx.x >> 5;     // 0..3  -> n-tile
    const int half  = lane >> 4;            // 0/1
    const int l16   = lane & 15;
    const int which = blockIdx.y;           // 0=Q 1=K 2=V

    const float* W   = (which == 0) ? Wq : (which == 1) ? Wk : Wv;
    const float* bia = (which == 0) ? bq : (which == 1) ? bk : bv;
    float*       out = (which == 0) ? Q  : (which == 1) ? K  : V;

    const int col = wave * 16 + l16;        // B/C column for this lane

    // hoist all B fragments (16 k-steps) into registers: 32 VGPRs
    v2f bf[16];
#pragma unroll
    for (int kk = 0; kk < 16; ++kk) {
        const int ka = kk * 4 + half * 2;
        bf[kk].x = W[(size_t)ka * HD + col];
        bf[kk].y = W[(size_t)(ka + 1) * HD + col];
    }
    const float bias = bia[col];

    for (int t = 0; t < NODE_TILES_PER_BLOCK; ++t) {
        const int mtile = blockIdx.x * NODE_TILES_PER_BLOCK + t;
        const int arow  = mtile * 16 + l16;
        v8f c = {};
#pragma unroll
        for (int kk = 0; kk < 16; ++kk) {
            const int ka = kk * 4 + half * 2;
            v2f a;
            a.x = x[(size_t)arow * IN_DIM + ka];
            a.y = x[(size_t)arow * IN_DIM + ka + 1];
            c = __builtin_amdgcn_wmma_f32_16x16x4_f32(false, a, false, bf[kk],
                                                      (short)0, c, false, false);
        }
#pragma unroll
        for (int v = 0; v < 8; ++v) {
            const int mo = mtile * 16 + v + half * 8;
            out[(size_t)mo * HD + col] = c[v] + bias;
        }
    }
}

// ---------------- kernel 2: edge projection (WMMA) + score + seg-max ----------------
// grid = n_edges/64 blocks; block = 256 (8 waves -> 8 col-tiles of 128); B in regs
__global__ void k_edge_gemm_score(const float* __restrict__ ea,
                                  const float* __restrict__ We,
                                  const float* __restrict__ be,
                                  const long long* __restrict__ eidx,
                                  const float* __restrict__ Q,
                                  const float* __restrict__ K,
                                  const float* __restrict__ Aw,
                                  float* __restrict__ wE,      // d_out edge region
                                  float* __restrict__ s_arr,   // (E, HEADS)
                                  float* __restrict__ m_arr,   // (N, HEADS)
                                  long long n_edges) {
    __shared__ float tile[16 * EDGE_TILES_PER_BLOCK][EHD + 4]; // E_proj (+bias)

    const int tid  = threadIdx.x;
    const int lane = tid & 31;
    const int wave = tid >> 5;              // 0..7
    const int half = lane >> 4;
    const int l16  = lane & 15;
    const long long e0 = (long long)blockIdx.x * (16 * EDGE_TILES_PER_BLOCK);

    const int col = wave * 16 + l16;        // 0..127

    // hoist B fragments for this wave's 16 columns of We: 32 VGPRs
    v2f bf[16];
#pragma unroll
    for (int kk = 0; kk < 16; ++kk) {
        const int ka = kk * 4 + half * 2;
        bf[kk].x = We[(size_t)ka * EHD + col];
        bf[kk].y = We[(size_t)(ka + 1) * EHD + col];
    }
    const float bias = be[col];

    for (int t = 0; t < EDGE_TILES_PER_BLOCK; ++t) {
        const long long arow = e0 + t * 16 + l16;
        v8f c = {};
#pragma unroll
        for (int kk = 0; kk < 16; ++kk) {
            const int ka = kk * 4 + half * 2;
            v2f a;
            a.x = ea[arow * IN_DIM + ka];
            a.y = ea[arow * IN_DIM + ka + 1];
            c = __builtin_amdgcn_wmma_f32_16x16x4_f32(false, a, false, bf[kk],
                                                      (short)0, c, false, false);
        }
#pragma unroll
        for (int v = 0; v < 8; ++v)
            tile[t * 16 + v + half * 8][col] = c[v] + bias;
    }
    __syncthreads();

    // 64 edges x 64 (h,d) = 4096 items; 256 threads x 16
#pragma unroll
    for (int it = 0; it < 16; ++it) {
        const int idx     = tid + it * 256;   // 0..4095
        const int e_local = idx >> 6;         // 0..63
        const int hd      = idx & 63;
        const int h       = hd >> 3;
        const int d       = hd & 7;
        const long long edge = e0 + e_local;
        const long long src  = eidx[edge];
        const long long dst  = eidx[n_edges + edge];

        const float ew = tile[e_local][h * 16 + d];
        const float eb = tile[e_local][h * 16 + 8 + d];
        float sc = (K[src * HD + hd] + Q[dst * HD + hd]) * ew;
        sc = sqrtf(fmaxf(sc, 0.0f) + EPS_V) - sqrtf(fmaxf(-sc, 0.0f) + EPS_V);
        const float et = sc + eb;
        wE[edge * HD + hd] = et;

        // s[e,h] = sum_d et * Aw[d,h]; 8 consecutive lanes share (e,h)
        float r = et * Aw[d * HEADS + h];
        r += __shfl_xor(r, 1, 8);
        r += __shfl_xor(r, 2, 8);
        r += __shfl_xor(r, 4, 8);
        if (d == 0) {
            const float s = fminf(fmaxf(r, -CLAMP_V), CLAMP_V);
            s_arr[edge * HEADS + h] = s;
            atomic_max_float(&m_arr[dst * HEADS + h], s);
        }
    }
}

// ---------------- kernel 3: exp + segment denom ----------------
__global__ void k_edge_exp(const long long* __restrict__ eidx,
                           const float* __restrict__ m_arr,
                           float* __restrict__ s_arr,     // in: s, out: e
                           float* __restrict__ denom,
                           long long n_edges) {
    const long long tid = (long long)blockIdx.x * blockDim.x + threadIdx.x;
    if (tid >= n_edges * HEADS) return;
    const long long edge = tid >> 3;
    const int h = (int)(tid & 7);
    const long long dst = eidx[n_edges + edge];
    const float e = expf(s_arr[tid] - m_arr[dst * HEADS + h]);
    s_arr[tid] = e;
    atomicAdd(&denom[dst * HEADS + h], e);
}

// ---------------- kernel 4: normalize + scatter-add ----------------
__global__ void k_edge_scatter(const long long* __restrict__ eidx,
                               const float* __restrict__ e_arr,
                               const float* __restrict__ denom,
                               const float* __restrict__ V,
                               const float* __restrict__ wE,
                               float* __restrict__ wV,     // d_out node region (accum)
                               float* __restrict__ rowV,
                               long long n_edges) {
    const long long tid = (long long)blockIdx.x * blockDim.x + threadIdx.x;
    if (tid >= n_edges * HD) return;
    const long long edge = tid >> 6;
    const int hd = (int)(tid & 63);
    const int h  = hd >> 3;
    const long long src = eidx[edge];
    const long long dst = eidx[n_edges + edge];
    const float attn = e_arr[edge * HEADS + h] / (denom[dst * HEADS + h] + 1e-16f);
    atomicAdd(&wV[dst * HD + hd], V[src * HD + hd] * attn);
    atomicAdd(&rowV[dst * HD + hd], wE[edge * HD + hd] * attn);
}

// ---------------- kernel 5: wV += rowV @ VeRow (per head 8x8) ----------------
__global__ void k_node_final(const float* __restrict__ rowV,
                             const float* __restrict__ VeRow,
                             float* __restrict__ wV, int n_nodes) {
    const int tid = blockIdx.x * blockDim.x + threadIdx.x;
    if (tid >= n_nodes * HD) return;
    const int n  = tid >> 6;
    const int hc = tid & 63;
    const int h  = hc >> 3;
    const int cc = hc & 7;
    float acc = wV[tid];
#pragma unroll
    for (int d = 0; d < 8; ++d)
        acc += rowV[(size_t)n * HD + h * OUT_DIM + d] *
               VeRow[(size_t)d * HD + h * OUT_DIM + cc];
    wV[tid] = acc;
}

// ---------------- launch ----------------
extern "C" void kernel_launch(void* const* d_in, const int* in_sizes, int n_in,
                              void* d_out, int out_size, void* d_ws, size_t ws_size,
                              hipStream_t stream) {
    const float* x     = (const float*)d_in[0];
    const float* ea    = (const float*)d_in[1];
    const long long* eidx = (const long long*)d_in[2];
    const float* Wq = (const float*)d_in[3];
    const float* bq = (const float*)d_in[4];
    const float* Wk = (const float*)d_in[5];
    const float* bk = (const float*)d_in[6];
    const float* We = (const float*)d_in[7];
    const float* be = (const float*)d_in[8];
    const float* Wv = (const float*)d_in[9];
    const float* bv = (const float*)d_in[10];
    const float* Aw = (const float*)d_in[11];
    const float* VeRow = (const float*)d_in[12];

    const int n_nodes = in_sizes[0] / IN_DIM;          // 50000
    const long long n_edges = in_sizes[1] / IN_DIM;    // 800000

    float* out = (float*)d_out;
    float* wV  = out;                                  // n_nodes*64
    float* wE  = out + (size_t)n_nodes * HD;           // n_edges*64

    float* ws = (float*)d_ws;
    size_t off = 0;
    float* Q     = ws + off; off += (size_t)n_nodes * HD;
    float* K     = ws + off; off += (size_t)n_nodes * HD;
    float* V     = ws + off; off += (size_t)n_nodes * HD;
    float* s_arr = ws + off; off += (size_t)n_edges * HEADS;
    float* m_arr = ws + off; off += (size_t)n_nodes * HEADS;
    float* denom = ws + off; off += (size_t)n_nodes * HEADS;
    float* rowV  = ws + off; off += (size_t)n_nodes * HD;
    (void)ws_size; (void)n_in; (void)out_size;

    {   // init accumulators (covers both HD- and HEADS-sized regions)
        const int total = n_nodes * HD;
        k_init<<<(total + 255) / 256, 256, 0, stream>>>(wV, m_arr, denom, rowV, n_nodes);
    }
    {   // Q,K,V projections: 3125 m-tiles = 5 per block * 625 blocks
        dim3 grid((n_nodes / 16) / NODE_TILES_PER_BLOCK, 3);
        k_node_gemm<<<grid, 128, 0, stream>>>(x, Wq, bq, Wk, bk, Wv, bv, Q, K, V);
    }
    {   // edge projection + scores + segment max: 64 edges per block
        dim3 grid((unsigned)(n_edges / (16 * EDGE_TILES_PER_BLOCK)));
        k_edge_gemm_score<<<grid, 256, 0, stream>>>(ea, We, be, eidx, Q, K, Aw,
                                                    wE, s_arr, m_arr, n_edges);
    }
    {   // exp + denom
        const long long total = n_edges * HEADS;
        k_edge_exp<<<(unsigned)((total + 255) / 256), 256, 0, stream>>>(
            eidx, m_arr, s_arr, denom, n_edges);
    }
    {   // normalize + scatter
        const long long total = n_edges * HD;
        k_edge_scatter<<<(unsigned)((total + 255) / 256), 256, 0, stream>>>(
            eidx, s_arr, denom, V, wE, wV, rowV, n_edges);
    }
    {   // final per-node einsum
        const int total = n_nodes * HD;
        k_node_final<<<(total + 255) / 256, 256, 0, stream>>>(rowV, VeRow, wV, n_nodes);
    }
}